// Interpolator_71201967833782
// MI455X (gfx1250) — compile-verified
//
#include <hip/hip_runtime.h>

// ---------------------------------------------------------------------------
// Fused polyphase resampler (UP=3/DOWN=2, 63-tap h) + 51-tap FIR.
// One kernel: x -> LDS -> u tile (VALU polyphase) -> LDS -> 16x16x4 FP32 WMMA
// banded-Toeplitz GEMM -> out. No intermediate global round-trip (saves
// ~100 MB of traffic vs. the two-kernel version; HBM floor ~84 MB @ 23.3 TB/s).
// ---------------------------------------------------------------------------

typedef __attribute__((ext_vector_type(2))) float v2f;
typedef __attribute__((ext_vector_type(8))) float v8f;

#define UPF        3
#define DOWNF      2

// stage-1 (polyphase) parameters
#define PH_TAPS    21                    // 63 taps / 3 phases
#define BIAS       192                   // multiple of 3; makes s+BIAS >= 0

// stage-2 (51-tap FIR as WMMA GEMM) parameters
#define NTAPS2     51
#define OFF        50                    // Lb-1 : A-window shift
#define KT         68                    // >= 51+15, multiple of 4 -> 17 steps
#define WAVE_OUT   256                   // one 16x16 tile per wave
#define NWAVES     8
#define BLOCK_T    (NWAVES * 32)         // 256 threads
#define BLOCK_OUT  (NWAVES * WAVE_OUT)   // 2048 outputs per block
#define UTILE      (BLOCK_OUT + KT - 16) // 2100 u values per block
#define UTILE_PAD  2112
#define XS_N       1421                  // x-window: ceil(2*2099/3)+1+20
#define XS_PAD     1424

// ---------------------------------------------------------------------------
__global__ __launch_bounds__(BLOCK_T)
void resample_fir_fused(const float* __restrict__ x,
                        const float* __restrict__ h,
                        const float* __restrict__ bq,
                        float* __restrict__ out,
                        const int n_in, const int n_u, const int n_out)
{
    __shared__ float xs[XS_PAD];        // zero-extended x window
    __shared__ float u_s[UTILE_PAD];    // resampled tile (incl. halo)
    __shared__ float B_s[KT * 16];      // banded tap matrix
    __shared__ float hs[64];            // resample taps

    const int blockBase = (int)blockIdx.x * BLOCK_OUT;
    const int t0 = blockBase - OFF;                 // first u index of tile
    // qStart = floor((2*t0+32)/3) - 20, floor-correct for t0 = -50 via bias
    const int s0     = 2 * t0 + 32;
    const int qStart = (s0 + BIAS) / 3 - (BIAS / 3) - 20;

    // ---- phase 0: stage x window, taps, and banded B matrix into LDS ----
    for (int i = threadIdx.x; i < XS_N; i += BLOCK_T) {
        const int qg = qStart + i;
        xs[i] = (qg >= 0 && qg < n_in) ? __ldg(x + qg) : 0.0f;
    }
    if (threadIdx.x < 63) hs[threadIdx.x] = h[threadIdx.x];
    for (int i = threadIdx.x; i < KT * 16; i += BLOCK_T) {
        const int k = i >> 4;
        const int n = i & 15;
        const int j = OFF + n - k;
        B_s[i] = (j >= 0 && j < NTAPS2) ? __ldg(bq + j) : 0.0f;
    }
    __syncthreads();

    // ---- phase 1: polyphase resample into u_s ----
    // u[t] = sum_{r=0..20} h[p+3r] * x[q0-r], p=(2t+32)%3, q0=(2t+32-p)/3
    for (int i = threadIdx.x; i < UTILE; i += BLOCK_T) {
        const int t  = t0 + i;
        const int sb = 2 * t + 32 + BIAS;           // >= 0 always
        const int p  = sb % 3;
        const int q0 = sb / 3 - (BIAS / 3);
        const int li = q0 - qStart;                 // in [20, 1420]
        float acc = 0.0f;
#pragma unroll
        for (int r = 0; r < PH_TAPS; ++r)
            acc = __builtin_fmaf(hs[p + 3 * r], xs[li - r], acc);
        u_s[i] = (t >= 0 && t < n_u) ? acc : 0.0f;  // u_ext is 0 outside [0,n_u)
    }
    __syncthreads();

    // ---- phase 2: 51-tap FIR as banded-Toeplitz GEMM on the matrix pipe ----
    // C[r,c] = out[base+16r+c] = sum_k A[r,k]*B[k,c]
    //   A[r,k] = u_ext[base+16r+k-50] = u_s[wave*256 + 16r + k]
    //   B[k,c] = b[50+c-k] (0 outside taps)
    // Packing (ISA 7.12.2, wave32): A/B lane L: row/col=L%16, VGPR v holds
    // K = 4*kk + 2*(L/16) + v.  C/D lane L: col=L%16, VGPR i row M=i+8*(L/16).
    const int wave = threadIdx.x >> 5;
    const int lane = threadIdx.x & 31;
    const int nm   = lane & 15;
    const int hi   = lane >> 4;

    v8f acc = {};
    const int aBase = wave * WAVE_OUT + 16 * nm;

#pragma unroll
    for (int kk = 0; kk < KT / 4; ++kk) {
        const int k0 = kk * 4 + 2 * hi;
        v2f a, bb;
        a.x  = u_s[aBase + k0];
        a.y  = u_s[aBase + k0 + 1];
        bb.x = B_s[(k0    ) * 16 + nm];
        bb.y = B_s[(k0 + 1) * 16 + nm];
        // (neg_a, A, neg_b, B, c_mod, C, reuse_a, reuse_b)
        acc = __builtin_amdgcn_wmma_f32_16x16x4_f32(false, a, false, bb,
                                                    (short)0, acc, false, false);
    }

    const int outBase = blockBase + wave * WAVE_OUT;
#pragma unroll
    for (int i = 0; i < 8; ++i) {
        const int oi = outBase + 16 * (i + 8 * hi) + nm;
        if (oi < n_out) out[oi] = acc[i];
    }
}

// ---------------------------------------------------------------------------
extern "C" void kernel_launch(void* const* d_in, const int* in_sizes, int n_args,
                              void* d_out, int out_size, void* d_ws, size_t ws_size,
                              hipStream_t stream)
{
    (void)n_args; (void)d_ws; (void)ws_size;

    const float* x   = (const float*)d_in[0];   // (8, 1048576) f32, flat
    const float* h   = (const float*)d_in[1];   // 63-tap resample filter
    const float* bfi = (const float*)d_in[2];   // 51-tap FIR
    float*       out = (float*)d_out;

    const int n_in = in_sizes[0];
    const long long prod = (long long)n_in * UPF;
    const int n_u  = (int)(prod / DOWNF + ((prod % DOWNF) ? 1 : 0)); // 12,582,912
    const int nout = out_size;                                       // 12,583,008

    const int grid = (nout + BLOCK_OUT - 1) / BLOCK_OUT;             // 6145
    resample_fir_fused<<<grid, BLOCK_T, 0, stream>>>(x, h, bfi, out,
                                                     n_in, n_u, nout);
}